// TemporalGCN_1288490189101
// MI455X (gfx1250) — compile-verified
//
#include <hip/hip_runtime.h>

#define NN 100000
#define NE 1600000
#define NG 64
#define DD 64

typedef float v2f __attribute__((ext_vector_type(2)));
typedef float v8f __attribute__((ext_vector_type(8)));

// ---------------- degree / normalization ----------------

__global__ __launch_bounds__(256) void k_init_deg(float* __restrict__ deg, int n) {
  int i = blockIdx.x * blockDim.x + threadIdx.x;
  if (i < n) deg[i] = 1.0f;  // self-loop
}

__global__ __launch_bounds__(256) void k_deg_edges(const int* __restrict__ dst,
                                                   float* __restrict__ deg, int e) {
  int i = blockIdx.x * blockDim.x + threadIdx.x;
  if (i < e) unsafeAtomicAdd(&deg[dst[i]], 1.0f);
}

__global__ __launch_bounds__(256) void k_dinv(float* __restrict__ deg, int n) {
  int i = blockIdx.x * blockDim.x + threadIdx.x;
  if (i < n) deg[i] = rsqrtf(fmaxf(deg[i], 1.0f));
}

// ---------------- fp32 WMMA GEMM: Y[M,64] = X[M,64] @ W[64,64] (+bias, relu) ----
// One 128-thread block (4 wave32) per 16-row tile; wave w owns cols [16w,16w+16).
// V_WMMA_F32_16X16X4_F32 VGPR layouts per CDNA5 ISA 7.12.2.

__global__ __launch_bounds__(128) void k_gemm64(const float* __restrict__ X,
                                                const float* __restrict__ W,
                                                const float* __restrict__ bias,
                                                float* __restrict__ Y,
                                                int M, int relu) {
  const int lane    = threadIdx.x & 31;
  const int wave    = threadIdx.x >> 5;
  const int rowBase = blockIdx.x * 16;
  const int colBase = wave * 16;
  const int r16     = lane & 15;
  const int kb      = (lane >> 4) * 2;  // lanes 0-15: K+0/K+1; lanes 16-31: K+2/K+3

  const float* xrow = X + (size_t)(rowBase + r16) * DD;

  v8f c = {};
#pragma unroll
  for (int kk = 0; kk < DD; kk += 4) {
    v2f a;
    a.x = xrow[kk + kb];
    a.y = xrow[kk + kb + 1];
    v2f b;
    b.x = W[(size_t)(kk + kb) * DD + colBase + r16];
    b.y = W[(size_t)(kk + kb + 1) * DD + colBase + r16];
    c = __builtin_amdgcn_wmma_f32_16x16x4_f32(false, a, false, b,
                                              (short)0, c, false, false);
  }

  const int rowOff = (lane >> 4) * 8;  // lanes 16-31 hold M = r+8
  const int col    = colBase + r16;
  const float bv   = bias ? bias[col] : 0.0f;
#pragma unroll
  for (int r = 0; r < 8; ++r) {
    float v = c[r] + bv;
    if (relu) v = fmaxf(v, 0.0f);
    Y[(size_t)(rowBase + rowOff + r) * DD + col] = v;
  }
}

// ---------------- edge gather + scale + scatter-add ----------------
// 16 threads/edge, float4 per thread; hardware f32 atomics into L2-resident acc.

__global__ __launch_bounds__(256) void k_edge_scatter(const int* __restrict__ src,
                                                      const int* __restrict__ dst,
                                                      const float* __restrict__ dinv,
                                                      const float* __restrict__ xw,
                                                      float* __restrict__ acc, int e) {
  int idx = blockIdx.x * blockDim.x + threadIdx.x;
  if (idx >= e * 16) return;
  int ei = idx >> 4;
  int f4 = (idx & 15) << 2;
  int s = src[ei], d = dst[ei];
  float cf = dinv[s] * dinv[d];
  const float4 v = *(const float4*)(xw + (size_t)s * DD + f4);
  float* o = acc + (size_t)d * DD + f4;
  unsafeAtomicAdd(o + 0, v.x * cf);
  unsafeAtomicAdd(o + 1, v.y * cf);
  unsafeAtomicAdd(o + 2, v.z * cf);
  unsafeAtomicAdd(o + 3, v.w * cf);
}

// acc = relu(acc + xw*dinv^2 (self-loop) + bias)
__global__ __launch_bounds__(256) void k_finalize(float* __restrict__ acc,
                                                  const float* __restrict__ xw,
                                                  const float* __restrict__ dinv,
                                                  const float* __restrict__ bias,
                                                  int n) {
  int idx = blockIdx.x * blockDim.x + threadIdx.x;
  if (idx >= n * DD) return;
  int i = idx >> 6, f = idx & 63;
  float di = dinv[i];
  float v = acc[idx] + xw[idx] * di * di + bias[f];
  acc[idx] = fmaxf(v, 0.0f);
}

// ---------------- global mean pool ----------------

__global__ __launch_bounds__(256) void k_pool(const float* __restrict__ h,
                                              const int* __restrict__ batch,
                                              float* __restrict__ gsum,
                                              float* __restrict__ cnt, int n) {
  int idx = blockIdx.x * blockDim.x + threadIdx.x;
  if (idx >= n * 16) return;
  int i = idx >> 4, f4 = (idx & 15) << 2;
  int g = batch[i];
  const float4 v = *(const float4*)(h + (size_t)i * DD + f4);
  float* o = gsum + (size_t)g * DD + f4;
  unsafeAtomicAdd(o + 0, v.x);
  unsafeAtomicAdd(o + 1, v.y);
  unsafeAtomicAdd(o + 2, v.z);
  unsafeAtomicAdd(o + 3, v.w);
  if (f4 == 0) unsafeAtomicAdd(&cnt[g], 1.0f);
}

__global__ __launch_bounds__(256) void k_pool_div(float* __restrict__ gsum,
                                                  const float* __restrict__ cnt) {
  int idx = blockIdx.x * blockDim.x + threadIdx.x;
  if (idx >= NG * DD) return;
  gsum[idx] = gsum[idx] / fmaxf(cnt[idx >> 6], 1.0f);
}

// ---------------- driver ----------------

extern "C" void kernel_launch(void* const* d_in, const int* in_sizes, int n_in,
                              void* d_out, int out_size, void* d_ws, size_t ws_size,
                              hipStream_t stream) {
  const float* x    = (const float*)d_in[0];
  const int*   ei   = (const int*)d_in[1];
  const int*   batch= (const int*)d_in[2];
  const float* W1   = (const float*)d_in[3];
  const float* b1   = (const float*)d_in[4];
  const float* W2   = (const float*)d_in[5];
  const float* b2   = (const float*)d_in[6];
  const float* Wp   = (const float*)d_in[7];
  const float* bp   = (const float*)d_in[8];
  float* out = (float*)d_out;

  const int* srcv = ei;
  const int* dstv = ei + NE;

  // workspace layout (floats)
  float* A    = (float*)d_ws;              // [NN*64] xw scratch
  float* B    = A + (size_t)NN * DD;       // [NN*64] acc / hidden
  float* dinv = B + (size_t)NN * DD;       // [NN]
  float* gsum = dinv + NN;                 // [NG*64]
  float* cnt  = gsum + (size_t)NG * DD;    // [NG]

  const int TB = 256;
  dim3 b256(TB), b128(128);

  // degrees -> dinv
  k_init_deg<<<(NN + TB - 1) / TB, b256, 0, stream>>>(dinv, NN);
  k_deg_edges<<<(NE + TB - 1) / TB, b256, 0, stream>>>(dstv, dinv, NE);
  k_dinv<<<(NN + TB - 1) / TB, b256, 0, stream>>>(dinv, NN);

  // ---- layer 1 ----
  k_gemm64<<<NN / 16, b128, 0, stream>>>(x, W1, nullptr, A, NN, 0);
  hipMemsetAsync(B, 0, (size_t)NN * DD * sizeof(float), stream);
  k_edge_scatter<<<(NE * 16 + TB - 1) / TB, b256, 0, stream>>>(srcv, dstv, dinv, A, B, NE);
  k_finalize<<<(NN * DD + TB - 1) / TB, b256, 0, stream>>>(B, A, dinv, b1, NN);

  // ---- layer 2 (ping-pong: xw2 in A, acc back into B) ----
  k_gemm64<<<NN / 16, b128, 0, stream>>>(B, W2, nullptr, A, NN, 0);
  hipMemsetAsync(B, 0, (size_t)NN * DD * sizeof(float), stream);
  k_edge_scatter<<<(NE * 16 + TB - 1) / TB, b256, 0, stream>>>(srcv, dstv, dinv, A, B, NE);
  k_finalize<<<(NN * DD + TB - 1) / TB, b256, 0, stream>>>(B, A, dinv, b2, NN);

  // ---- mean pool ----
  hipMemsetAsync(gsum, 0, (size_t)NG * DD * sizeof(float), stream);
  hipMemsetAsync(cnt, 0, (size_t)NG * sizeof(float), stream);
  k_pool<<<(NN * 16 + TB - 1) / TB, b256, 0, stream>>>(B, batch, gsum, cnt, NN);
  k_pool_div<<<(NG * DD + TB - 1) / TB, b256, 0, stream>>>(gsum, cnt);

  // ---- head GEMM: out = pooled @ Wp + bp ----
  k_gemm64<<<NG / 16, b128, 0, stream>>>(gsum, Wp, bp, out, NG, 0);

  (void)in_sizes; (void)n_in; (void)out_size; (void)ws_size;
}